// DetectionSegmentationConsistency_43619687858416
// MI455X (gfx1250) — compile-verified
//
#include <hip/hip_runtime.h>

// ---------------------------------------------------------------------------
// DetectionSegmentationConsistency for MI455X (gfx1250, wave32)
//   d[b,y,x] = seg[b,2,y,x] - seg[b,1,y,x]
//   K1: TDM (tensor_load_to_lds) streams 16x128 strips of both planes into
//       LDS; each wave computes a 16x16 in-tile inclusive column prefix via
//       4x V_WMMA_F32_16X16X4_F32 (scan == lower-triangular-ones matmul).
//   K2: exclusive prefix of 16-row block sums  -> C[b,yb,x]  (8.5 MB, L2)
//   K3: per-box gather/reduce + atomicAdd of relu((f-d)/area)*conf/(B*N)
// Workspace: L = 32*1024*1024 f32 (128 MiB), C = 32*65*1024 f32 (8.1 MiB)
// ---------------------------------------------------------------------------

typedef float v2f __attribute__((ext_vector_type(2)));
typedef float v8f __attribute__((ext_vector_type(8)));
typedef unsigned int u32x4 __attribute__((ext_vector_type(4)));
typedef int i32x4 __attribute__((ext_vector_type(4)));
typedef int i32x8 __attribute__((ext_vector_type(8)));

constexpr int BB = 32;
constexpr int NN = 300;
constexpr int HH = 1024;
constexpr int WW = 1024;
constexpr float CONF_THRESH = 0.3f;

// ---------------- K0: zero the scalar output --------------------------------
__global__ void k0_zero(float* __restrict__ out, int n) {
  int i = blockIdx.x * blockDim.x + threadIdx.x;
  if (i < n) out[i] = 0.0f;
}

#if __has_builtin(__builtin_amdgcn_tensor_load_to_lds)
#define USE_TDM 1
#else
#define USE_TDM 0
#endif

#if USE_TDM
// Issue one TDM load of a 16x128 f32 tile (row stride 1024 elems) into LDS.
// D# layout per CDNA5 ISA ch.8: group0 = {ctrl, lds_addr, global_addr, type},
// group1 = {mask/data_size, tensor dims, tile dims, strides}.
__device__ __forceinline__ void tdm_load_tile_16x128(const float* gsrc,
                                                     unsigned lds_off) {
  unsigned long long ga = (unsigned long long)(uintptr_t)gsrc;
  u32x4 g0;
  g0[0] = 1u;                                        // count=1 (user mode)
  g0[1] = lds_off;                                   // lds_addr (bytes)
  g0[2] = (unsigned)(ga & 0xffffffffu);              // global_addr[31:0]
  g0[3] = (unsigned)((ga >> 32) & 0x01ffffffu)       // global_addr[56:32]
          | 0x80000000u;                             // type=2 ("image")
  i32x8 g1;
  g1[0] = (int)(2u << 16);            // workgroup_mask=0, data_size=2 (4B)
  g1[1] = (int)(1024u << 16);         // tensor_dim0[15:0]=1024 at bits[79:64]
  g1[2] = (int)(1024u << 16);         // tensor_dim0 hi=0 | tensor_dim1 lo=1024
  g1[3] = (int)(128u << 16);          // tensor_dim1 hi=0 | tile_dim0=128
  g1[4] = (int)16u;                   // tile_dim1=16, tile_dim2=0
  g1[5] = (int)1024u;                 // tensor_dim0_stride[31:0] = 1024 elems
  g1[6] = 0;                          // stride0 hi | dim1_stride lo16 = 0
  g1[7] = 0x10;                       // dim1_stride[47:16] -> 1<<20 elems
  i32x4 gz = {0, 0, 0, 0};            // groups 2/3 unused (2D tensor)
#if __clang_major__ >= 23
  i32x8 gz8 = {0, 0, 0, 0, 0, 0, 0, 0};
  __builtin_amdgcn_tensor_load_to_lds(g0, g1, gz, gz, gz8, 0);
#else
  __builtin_amdgcn_tensor_load_to_lds(g0, g1, gz, gz, 0);
#endif
}
#endif

// ---------------- K1: TDM -> LDS -> WMMA column-prefix ----------------------
// One 256-thread block per (b, ytile, 128-col strip); 8 waves, one 16x16
// sub-tile each. D(16x16) = sum_{c=0..3} A_c(16x4) x B_c(4x16), A_c
// triangular-ones chunks => D[m,n] = sum_{k<=m} d[k,n].
__global__ __launch_bounds__(256) void k1_tile_prefix(
    const float* __restrict__ seg, float* __restrict__ Lbuf) {
  __shared__ float lds1[16 * 128];   // driveway tile
  __shared__ float lds2[16 * 128];   // footpath tile

  const int wave = threadIdx.x >> 5;
  const int lane = threadIdx.x & 31;
  const int g = blockIdx.x;                  // 0 .. 32*64*8-1
  const int b  = g >> 9;                     // / (64 ytiles * 8 strips)
  const int t  = g & 511;
  const int y0 = (t >> 3) << 4;              // ytile * 16
  const int xs = (t & 7) << 7;               // strip start column (128 wide)

  const float* __restrict__ p1 = seg + (((size_t)b * 3 + 1) << 20); // driveway
  const float* __restrict__ p2 = seg + (((size_t)b * 3 + 2) << 20); // footpath

#if USE_TDM
  if (wave == 0) {
    tdm_load_tile_16x128(p1 + ((size_t)y0 << 10) + xs,
                         (unsigned)(uintptr_t)(void*)lds1);
    tdm_load_tile_16x128(p2 + ((size_t)y0 << 10) + xs,
                         (unsigned)(uintptr_t)(void*)lds2);
    __builtin_amdgcn_s_wait_tensorcnt(0);
  }
#else
  for (int i = threadIdx.x; i < 16 * 128; i += 256) {
    int r = i >> 7, cc = i & 127;
    size_t go = ((size_t)(y0 + r) << 10) + (size_t)(xs + cc);
    lds1[i] = p1[go];
    lds2[i] = p2[go];
  }
#endif
  __syncthreads();

  // gfx1250 prefetch of the next y-strip (global_prefetch_b8)
  const int half = lane >> 4;                // B/A operand K-pair selector
  const int colL = (wave << 4) + (lane & 15);  // column within strip
  if (y0 + 16 < HH) {
    size_t poff = ((size_t)(y0 + 16 + half) << 10) + (size_t)(xs + colL);
    __builtin_prefetch(p1 + poff, 0, 1);
    __builtin_prefetch(p2 + poff, 0, 1);
  }

  const float m = (float)(lane & 15);        // A-matrix row for this lane
  v8f acc = {};
#pragma unroll
  for (int c = 0; c < 4; ++c) {
    const int kx = 4 * c + 2 * half;         // K indices {kx, kx+1}
    v2f bmat;
    bmat.x = lds2[kx * 128 + colL] - lds1[kx * 128 + colL];
    bmat.y = lds2[(kx + 1) * 128 + colL] - lds1[(kx + 1) * 128 + colL];
    v2f amat;                                // triangular ones T[m,k] = (k<=m)
    amat.x = ((float)kx       <= m) ? 1.0f : 0.0f;
    amat.y = ((float)(kx + 1) <= m) ? 1.0f : 0.0f;
    acc = __builtin_amdgcn_wmma_f32_16x16x4_f32(
        /*neg_a=*/false, amat, /*neg_b=*/false, bmat,
        /*c_mod=*/(short)0, acc, /*reuse_a=*/false, /*reuse_b=*/false);
  }

  // C/D layout: VGPR r -> row r (lanes 0-15) / row r+8 (lanes 16-31)
  const int colg = xs + colL;
#pragma unroll
  for (int r = 0; r < 8; ++r) {
    const int row = y0 + r + 8 * half;
    Lbuf[(((size_t)b << 10) + (size_t)row) * (size_t)WW + (size_t)colg] = acc[r];
  }
}

// ---------------- K2: exclusive prefix of 16-row block sums -----------------
// Block sum of block yb == L[b, 16*yb+15, x] (row 15 of in-tile prefix).
__global__ __launch_bounds__(256) void k2_block_scan(
    const float* __restrict__ Lbuf, float* __restrict__ Cbuf) {
  const int tid = blockIdx.x * 256 + threadIdx.x;   // 0 .. BB*WW-1
  const int b = tid >> 10;
  const int x = tid & 1023;
  const float* __restrict__ lb = Lbuf + ((size_t)b << 20);
  float* __restrict__ cb = Cbuf + (size_t)b * 65 * (size_t)WW;
  float run = 0.0f;
  for (int yb = 0; yb < 64; ++yb) {
    cb[(size_t)yb * WW + x] = run;
    run += lb[((size_t)(yb * 16 + 15) << 10) + (size_t)x];
  }
  cb[(size_t)64 * WW + x] = run;
}

// ---------------- block reduction (wave32-aware) ----------------------------
__device__ __forceinline__ float blockReduceSum256(float v) {
  __shared__ float red[8];
  const int lane = threadIdx.x & 31;
  const int w    = threadIdx.x >> 5;
#pragma unroll
  for (int off = 16; off > 0; off >>= 1) v += __shfl_down(v, off, 32);
  if (lane == 0) red[w] = v;
  __syncthreads();
  if (w == 0) {
    v = (lane < 8) ? red[lane] : 0.0f;
#pragma unroll
    for (int off = 4; off > 0; off >>= 1) v += __shfl_down(v, off, 32);
  }
  return v;  // valid in thread 0
}

// ---------------- K3: per-box rectangle sums --------------------------------
__global__ __launch_bounds__(256) void k3_boxes(
    const float* __restrict__ boxes, const float* __restrict__ conf,
    const float* __restrict__ Lbuf, const float* __restrict__ Cbuf,
    float* __restrict__ out) {
  const int box = blockIdx.x;           // 0 .. BB*NN-1
  const int b = box / NN;

  const float cx = boxes[(size_t)box * 4 + 0];
  const float cy = boxes[(size_t)box * 4 + 1];
  const float bw = boxes[(size_t)box * 4 + 2];
  const float bh = boxes[(size_t)box * 4 + 3];
  const float cf = conf[box];

  // (int) cast == trunc toward zero, matching jnp.trunc().astype(int32)
  const int x1 = min(max((int)((cx - bw * 0.5f) * (float)WW), 0), WW - 1);
  const int x2 = min(max((int)((cx + bw * 0.5f) * (float)WW), 0), WW - 1);
  const int y1 = min(max((int)((cy - bh * 0.5f) * (float)HH), 0), HH - 1);
  const int y2 = min(max((int)((cy + bh * 0.5f) * (float)HH), 0), HH - 1);

  if (!(cf >= CONF_THRESH && x2 > x1 && y2 > y1)) return;  // uniform branch

  const float* __restrict__ lb = Lbuf + ((size_t)b << 20);
  const float* __restrict__ cb = Cbuf + (size_t)b * 65 * (size_t)WW;
  // exclusive column prefix at row y: P[y,x] = C[y>>4, x] + (y&15 ? L[y-1,x] : 0)
  const float* __restrict__ c2 = cb + (size_t)(y2 >> 4) * WW;
  const float* __restrict__ c1 = cb + (size_t)(y1 >> 4) * WW;
  const bool has2 = (y2 & 15) != 0;
  const bool has1 = (y1 & 15) != 0;
  const float* __restrict__ l2 = has2 ? (lb + ((size_t)(y2 - 1) << 10)) : lb;
  const float* __restrict__ l1 = has1 ? (lb + ((size_t)(y1 - 1) << 10)) : lb;

  float s = 0.0f;
  for (int x = x1 + (int)threadIdx.x; x < x2; x += 256) {
    float pv2 = c2[x] + (has2 ? l2[x] : 0.0f);
    float pv1 = c1[x] + (has1 ? l1[x] : 0.0f);
    s += pv2 - pv1;
  }
  s = blockReduceSum256(s);
  if (threadIdx.x == 0) {
    const float area = (float)((y2 - y1) * (x2 - x1));
    const float v = fmaxf(s / area, 0.0f) * cf * (1.0f / (float)(BB * NN));
    atomicAdd(out, v);
  }
}

// ---------------------------------------------------------------------------
extern "C" void kernel_launch(void* const* d_in, const int* in_sizes, int n_in,
                              void* d_out, int out_size, void* d_ws, size_t ws_size,
                              hipStream_t stream) {
  const float* boxes = (const float*)d_in[0];   // (32, 300, 4)
  const float* conf  = (const float*)d_in[1];   // (32, 300)
  const float* seg   = (const float*)d_in[2];   // (32, 3, 1024, 1024)
  float* out = (float*)d_out;

  float* Lbuf = (float*)d_ws;                             // 32*1024*1024 f32
  float* Cbuf = Lbuf + (size_t)BB * HH * (size_t)WW;      // 32*65*1024  f32
  // required ws: (33554432 + 2129920) * 4 bytes ~= 142.7 MB
  (void)ws_size; (void)in_sizes; (void)n_in;

  k0_zero<<<1, 32, 0, stream>>>(out, out_size);
  k1_tile_prefix<<<BB * 64 * 8, 256, 0, stream>>>(seg, Lbuf);
  k2_block_scan<<<(BB * WW) / 256, 256, 0, stream>>>(Lbuf, Cbuf);
  k3_boxes<<<BB * NN, 256, 0, stream>>>(boxes, conf, Lbuf, Cbuf, out);
}